// GraphAttentionEncoder_31662498906456
// MI455X (gfx1250) — compile-verified
//
#include <hip/hip_runtime.h>
#include <hip/hip_bf16.h>
#include <math.h>

typedef __attribute__((ext_vector_type(2))) float v2f;
typedef __attribute__((ext_vector_type(8))) float v8f;

#define NEG_SLOPE 0.2f

// ---------------------------------------------------------------------------
// util kernels
// ---------------------------------------------------------------------------
__global__ void k_fill(float* __restrict__ p, float v, int n) {
    int i = blockIdx.x * blockDim.x + threadIdx.x;
    if (i < n) p[i] = v;
}

// ---------------------------------------------------------------------------
// fp32 WMMA GEMM with LDS-staged B panel.
//   C[M,Ncol] = A[M,K] * B[K,Ncol], K processed in 128-row chunks.
//   Block = 256 threads = 8 waves; all waves share one (16*NT)-wide N panel
//   (B loaded from global once per block, consumed from LDS), each wave owns
//   one 16-row M tile with NT independent WMMA accumulator chains.
//   LDS row stride = COLS+8 so 2*stride % 64 == 16: the two lane-halves of a
//   ds_load hit disjoint 16-bank groups -> conflict-free.
// ---------------------------------------------------------------------------
template <int NT>
__global__ void k_gemm_wmma_lds(const float* __restrict__ A,
                                const float* __restrict__ B,
                                float* __restrict__ C,
                                int M, int K, int Ncol) {
    constexpr int COLS = 16 * NT;
    constexpr int LSTR = COLS + 8;
    constexpr int KC   = 128;
    __shared__ float sB[KC * LSTR];

    const int ngrp   = Ncol / COLS;
    const int gm     = blockIdx.x / ngrp;        // super-block of 8 M tiles
    const int gn     = blockIdx.x % ngrp;
    const int tn0    = gn * COLS;
    const int mtiles = M >> 4;

    const int lane  = threadIdx.x & 31;
    const int wib   = threadIdx.x >> 5;
    const int mt    = gm * 8 + wib;              // this wave's M tile
    const bool valid = (mt < mtiles);            // wave-uniform

    const int tm  = mt << 4;
    const int l15 = lane & 15;
    const int hi  = lane >> 4;                   // 0 or 1 (lane half)

    const v8f zero = {0.f, 0.f, 0.f, 0.f, 0.f, 0.f, 0.f, 0.f};
    v8f c[NT];
#pragma unroll
    for (int t = 0; t < NT; ++t) c[t] = zero;

    // A fragment: M = tm + l15, K = k + 2*hi + {0,1} (contiguous float2)
    const float* arow = A + (long)(tm + l15) * K + 2 * hi;
    // B fragment base within LDS chunk
    const float* sb = sB + l15 + 2 * hi * LSTR;

    const int col   = threadIdx.x % COLS;        // coop-load mapping
    const int r0    = threadIdx.x / COLS;
    const int rstep = blockDim.x / COLS;

    for (int kc = 0; kc < K; kc += KC) {
        // ---- cooperative load of B[kc..kc+KC) x COLS into LDS ----
        for (int r = r0; r < KC; r += rstep) {
            sB[r * LSTR + col] = B[(long)(kc + r) * Ncol + tn0 + col];
        }
        __syncthreads();

        if (valid) {
#pragma unroll 4
            for (int k0 = 0; k0 < KC; k0 += 4) {
                v2f a;
                a.x = arow[kc + k0 + 0];
                a.y = arow[kc + k0 + 1];
                const float* b0 = sb + k0 * LSTR;
#pragma unroll
                for (int t = 0; t < NT; ++t) {
                    v2f b;
                    b.x = b0[t * 16];
                    b.y = b0[t * 16 + LSTR];
                    c[t] = __builtin_amdgcn_wmma_f32_16x16x4_f32(
                               /*neg_a=*/false, a, /*neg_b=*/false, b,
                               /*c_mod=*/(short)0, c[t],
                               /*reuse_a=*/false, /*reuse_b=*/false);
                }
            }
        }
        __syncthreads();
    }

    if (valid) {
        // C/D layout: VGPR v holds row M = tm + v + 8*hi, col tn0 + 16*t + l15
#pragma unroll
        for (int t = 0; t < NT; ++t) {
#pragma unroll
            for (int v = 0; v < 8; ++v) {
                C[(long)(tm + v + 8 * hi) * Ncol + tn0 + t * 16 + l15] = c[t][v];
            }
        }
    }
}

// ---------------------------------------------------------------------------
// per-node attention logits: asrc[n,h] = <h[n,h,:], att_src[h,:]>, same for dst
// h laid out [N, H*C]; i = n*H + h indexes contiguous C-blocks.
// ---------------------------------------------------------------------------
__global__ void k_alpha_node(const float* __restrict__ h,
                             const float* __restrict__ att_src,
                             const float* __restrict__ att_dst,
                             float* __restrict__ asrc,
                             float* __restrict__ adst,
                             int NH, int H, int C) {
    int i = blockIdx.x * blockDim.x + threadIdx.x;
    if (i >= NH) return;
    int hh = i % H;
    const float* row = h + (long)i * C;
    const float* ws  = att_src + hh * C;
    const float* wd  = att_dst + hh * C;
    float s0 = 0.f, s1 = 0.f;
    for (int c = 0; c < C; ++c) {
        float v = row[c];
        s0 += v * ws[c];
        s1 += v * wd[c];
    }
    asrc[i] = s0;
    adst[i] = s1;
}

// ---------------------------------------------------------------------------
// edge helpers
// ---------------------------------------------------------------------------
__device__ __forceinline__ void edge_sd(const int* __restrict__ srcA,
                                        const int* __restrict__ dstA,
                                        int e, int E, int& s, int& d) {
    if (e < E) { s = srcA[e]; d = dstA[e]; }
    else       { s = e - E;   d = e - E;   }     // self loop
}

__device__ __forceinline__ float leaky(float v) {
    return v > 0.f ? v : NEG_SLOPE * v;
}

__device__ __forceinline__ void atomicMaxF32(float* addr, float val) {
    int* ia = (int*)addr;
    int old = __float_as_int(*addr);
    while (__int_as_float(old) < val) {
        int assumed = old;
        old = atomicCAS(ia, assumed, __float_as_int(val));
        if (old == assumed) break;
    }
}

// pass 1: segment max over dst
__global__ void k_edge_max(const int* __restrict__ srcA, const int* __restrict__ dstA,
                           const float* __restrict__ asrc, const float* __restrict__ adst,
                           float* __restrict__ m, int E, int Etot, int H) {
    int i = blockIdx.x * blockDim.x + threadIdx.x;
    if (i >= Etot * H) return;
    int e = i / H, hh = i % H;
    int s, d; edge_sd(srcA, dstA, e, E, s, d);
    float v = leaky(asrc[s * H + hh] + adst[d * H + hh]);
    atomicMaxF32(&m[d * H + hh], v);
}

// pass 2: segment sum of exp(e - m[dst])
__global__ void k_edge_expsum(const int* __restrict__ srcA, const int* __restrict__ dstA,
                              const float* __restrict__ asrc, const float* __restrict__ adst,
                              const float* __restrict__ m, float* __restrict__ ssum,
                              int E, int Etot, int H) {
    int i = blockIdx.x * blockDim.x + threadIdx.x;
    if (i >= Etot * H) return;
    int e = i / H, hh = i % H;
    int s, d; edge_sd(srcA, dstA, e, E, s, d);
    float v  = leaky(asrc[s * H + hh] + adst[d * H + hh]);
    float ev = expf(v - m[d * H + hh]);
    atomicAdd(&ssum[d * H + hh], ev);
}

// pass 3: normalized alpha per (edge, head)
__global__ void k_edge_alpha(const int* __restrict__ srcA, const int* __restrict__ dstA,
                             const float* __restrict__ asrc, const float* __restrict__ adst,
                             const float* __restrict__ m, const float* __restrict__ ssum,
                             float* __restrict__ alpha, int E, int Etot, int H) {
    int i = blockIdx.x * blockDim.x + threadIdx.x;
    if (i >= Etot * H) return;
    int e = i / H, hh = i % H;
    int s, d; edge_sd(srcA, dstA, e, E, s, d);
    float v  = leaky(asrc[s * H + hh] + adst[d * H + hh]);
    float ev = expf(v - m[d * H + hh]);
    alpha[i] = ev / (ssum[d * H + hh] + 1e-16f);
}

// pass 4: msg = h[src] * alpha, segment-sum into acc[dst]. F = H*C divides 256,
// so threads of a block map to whole edges -> fully coalesced gather/scatter.
__global__ void k_aggregate(const int* __restrict__ srcA, const int* __restrict__ dstA,
                            const float* __restrict__ hfeat, const float* __restrict__ alpha,
                            float* __restrict__ acc, int E, int Etot, int H, int C) {
    const int F = H * C;
    long idx = (long)blockIdx.x * blockDim.x + threadIdx.x;
    long total = (long)Etot * F;
    if (idx >= total) return;
    int e = (int)(idx / F);
    int j = (int)(idx % F);
    int hh = j / C;
    int s, d; edge_sd(srcA, dstA, e, E, s, d);
    float a = alpha[(long)e * H + hh];
    atomicAdd(&acc[(long)d * F + j], hfeat[(long)s * F + j] * a);
}

__global__ void k_bias_relu(float* __restrict__ p, const float* __restrict__ b,
                            int n, int F) {
    int i = blockIdx.x * blockDim.x + threadIdx.x;
    if (i >= n) return;
    float v = p[i] + b[i % F];
    p[i] = v > 0.f ? v : 0.f;
}

__global__ void k_bias_out(const float* __restrict__ acc, const float* __restrict__ b,
                           float* __restrict__ out, int n, int F) {
    int i = blockIdx.x * blockDim.x + threadIdx.x;
    if (i >= n) return;
    out[i] = acc[i] + b[i % F];
}

// ---------------------------------------------------------------------------
// launcher
// ---------------------------------------------------------------------------
extern "C" void kernel_launch(void* const* d_in, const int* in_sizes, int n_in,
                              void* d_out, int out_size, void* d_ws, size_t ws_size,
                              hipStream_t stream) {
    const float* x        = (const float*)d_in[0];
    const int*   eidx     = (const int*)d_in[1];       // [2, E]
    const float* W1       = (const float*)d_in[2];
    const float* att_src1 = (const float*)d_in[3];
    const float* att_dst1 = (const float*)d_in[4];
    const float* b1       = (const float*)d_in[5];
    const float* W2       = (const float*)d_in[6];
    const float* att_src2 = (const float*)d_in[7];
    const float* att_dst2 = (const float*)d_in[8];
    const float* b2       = (const float*)d_in[9];

    const int F_IN = 256, D1 = 256, H1 = 4, C1 = 64, OUT = 32;
    const int N    = in_sizes[0] / F_IN;
    const int E    = in_sizes[1] / 2;
    const int Etot = E + N;                            // + self loops
    const int* srcA = eidx;
    const int* dstA = eidx + E;

    // workspace layout (floats)
    float* ws    = (float*)d_ws;
    float* h1    = ws;                    // N*D1
    float* acc1  = h1   + (long)N * D1;   // N*D1  (-> h1r after bias+relu)
    float* h2    = acc1 + (long)N * D1;   // N*OUT
    float* acc2  = h2   + (long)N * OUT;  // N*OUT
    float* as1   = acc2 + (long)N * OUT;  // N*H1
    float* ad1   = as1  + (long)N * H1;
    float* m1    = ad1  + (long)N * H1;
    float* s1    = m1   + (long)N * H1;
    float* as2   = s1   + (long)N * H1;   // N
    float* ad2   = as2  + N;
    float* m2    = ad2  + N;
    float* s2    = m2   + N;
    float* alp1  = s2   + N;              // Etot*H1
    float* alp2  = alp1 + (long)Etot * H1;// Etot

    const int TB = 256;
    auto blocks = [](long n, int tb) { return (int)((n + tb - 1) / tb); };

    // init
    k_fill<<<blocks((long)N * H1, TB), TB, 0, stream>>>(m1, -INFINITY, N * H1);
    k_fill<<<blocks((long)N * H1, TB), TB, 0, stream>>>(s1, 0.f, N * H1);
    k_fill<<<blocks((long)N * D1, TB), TB, 0, stream>>>(acc1, 0.f, N * D1);
    k_fill<<<blocks(N, TB), TB, 0, stream>>>(m2, -INFINITY, N);
    k_fill<<<blocks(N, TB), TB, 0, stream>>>(s2, 0.f, N);
    k_fill<<<blocks((long)N * OUT, TB), TB, 0, stream>>>(acc2, 0.f, N * OUT);

    // ---- layer 1 ----
    {
        // 8 M-tiles (128 rows) x 64 cols per block
        int mtiles = N / 16;
        int gridm  = (mtiles + 7) / 8;
        int ngrp   = D1 / 64;
        k_gemm_wmma_lds<4><<<gridm * ngrp, TB, 0, stream>>>(x, W1, h1, N, F_IN, D1);
    }
    k_alpha_node<<<blocks((long)N * H1, TB), TB, 0, stream>>>(
        h1, att_src1, att_dst1, as1, ad1, N * H1, H1, C1);
    k_edge_max<<<blocks((long)Etot * H1, TB), TB, 0, stream>>>(
        srcA, dstA, as1, ad1, m1, E, Etot, H1);
    k_edge_expsum<<<blocks((long)Etot * H1, TB), TB, 0, stream>>>(
        srcA, dstA, as1, ad1, m1, s1, E, Etot, H1);
    k_edge_alpha<<<blocks((long)Etot * H1, TB), TB, 0, stream>>>(
        srcA, dstA, as1, ad1, m1, s1, alp1, E, Etot, H1);
    k_aggregate<<<blocks((long)Etot * D1, TB), TB, 0, stream>>>(
        srcA, dstA, h1, alp1, acc1, E, Etot, H1, C1);
    k_bias_relu<<<blocks((long)N * D1, TB), TB, 0, stream>>>(acc1, b1, N * D1, D1);

    // ---- layer 2 ----
    {
        int mtiles = N / 16;
        int gridm  = (mtiles + 7) / 8;
        int ngrp   = OUT / 32;
        k_gemm_wmma_lds<2><<<gridm * ngrp, TB, 0, stream>>>(acc1, W2, h2, N, D1, OUT);
    }
    k_alpha_node<<<blocks(N, TB), TB, 0, stream>>>(
        h2, att_src2, att_dst2, as2, ad2, N, 1, OUT);
    k_edge_max<<<blocks(Etot, TB), TB, 0, stream>>>(
        srcA, dstA, as2, ad2, m2, E, Etot, 1);
    k_edge_expsum<<<blocks(Etot, TB), TB, 0, stream>>>(
        srcA, dstA, as2, ad2, m2, s2, E, Etot, 1);
    k_edge_alpha<<<blocks(Etot, TB), TB, 0, stream>>>(
        srcA, dstA, as2, ad2, m2, s2, alp2, E, Etot, 1);
    k_aggregate<<<blocks((long)Etot * OUT, TB), TB, 0, stream>>>(
        srcA, dstA, h2, alp2, acc2, E, Etot, 1, OUT);

    // mean over 1 head == identity; add bias, write output
    k_bias_out<<<blocks((long)N * OUT, TB), TB, 0, stream>>>(
        acc2, b2, (float*)d_out, N * OUT, OUT);
}